// SNNROE_9225589752281
// MI455X (gfx1250) — compile-verified
//
#include <hip/hip_runtime.h>
#include <math.h>

typedef __attribute__((ext_vector_type(2))) float v2f;
typedef __attribute__((ext_vector_type(8))) float v8f;

#define T_DIM       16
#define B_DIM       32
#define F_DIM       65536            // 64*32*32
#define BF          (B_DIM * F_DIM)  // stride between t-slices
#define TILE_F      128              // f-columns per wave tile
#define ROW_STRIDE  (TILE_F + 4)     // LDS row pad -> conflict-free b64 frag reads
#define WAVES       4
#define BLOCK       (WAVES * 32)
#define WGS_PER_B   64
#define TILES_PER_B (F_DIM / TILE_F)          // 512
#define TILE_STEP   (WGS_PER_B * WAVES)       // 256 -> 2 tiles per wave

// ---------------------------------------------------------------- zero ws
__global__ void SNNROE_zero(float* __restrict__ ws) {
    int i = blockIdx.x * blockDim.x + threadIdx.x;
    if (i < B_DIM * 256) ws[i] = 0.0f;
}

// ---------------------------------------------------------------- main pass
// Streams x once; accumulates per-b 16x16 Gram of r[t] via v_wmma_f32_16x16x4_f32.
__global__ __launch_bounds__(BLOCK) void SNNROE_gram(const float* __restrict__ x,
                                                     float* __restrict__ ws) {
    __shared__ float lds[WAVES * T_DIM * ROW_STRIDE];

    const int b    = blockIdx.x;
    const int wg   = blockIdx.y;
    const int lane = threadIdx.x & 31;
    const int wave = threadIdx.x >> 5;

    float*       wl = lds + wave * (T_DIM * ROW_STRIDE);
    const float* xb = x + (size_t)b * F_DIM;

    v8f acc0 = {};
    v8f acc1 = {};

    // A/B fragment read offset in the wave's LDS tile (16x4 f32 WMMA layout):
    // lane L -> row t = L%16, cols 2*(L/16)+{0,1}
    const int frag_off = (lane & 15) * ROW_STRIDE + 2 * (lane >> 4);

    for (int tile = wg * WAVES + wave; tile < TILES_PER_B; tile += TILE_STEP) {
        const int f0 = tile * TILE_F + lane * 4;

        // Prefetch the next tile this wave will touch (speculative, dropped if OOB).
        if (tile + TILE_STEP < TILES_PER_B) {
            __builtin_prefetch(xb + (size_t)(tile + TILE_STEP) * TILE_F + lane * 4, 0, 0);
        }

        // ---- stream 16 t-slices: relu + running prefix sum, r[t] -> LDS
        float s0 = 0.f, s1 = 0.f, s2 = 0.f, s3 = 0.f;
#pragma unroll
        for (int t = 0; t < T_DIM; ++t) {
            const float4 v = *(const float4*)(xb + (size_t)t * BF + f0);
            s0 += fmaxf(v.x, 0.f);
            s1 += fmaxf(v.y, 0.f);
            s2 += fmaxf(v.z, 0.f);
            s3 += fmaxf(v.w, 0.f);
            const float inv = 1.0f / (float)(t + 1);
            *(float4*)(wl + t * ROW_STRIDE + lane * 4) =
                make_float4(s0 * inv, s1 * inv, s2 * inv, s3 * inv);
        }
        __syncthreads();

        // ---- Gram accumulation: G += A * A^T, K in chunks of 4.
        // Symmetric Gram => identical per-lane data for A and B fragments.
#pragma unroll
        for (int kc = 0; kc < TILE_F / 4; kc += 2) {
            v2f fa = *(const v2f*)(wl + frag_off + kc * 4);
            acc0 = __builtin_amdgcn_wmma_f32_16x16x4_f32(
                false, fa, false, fa, (short)0, acc0, false, false);
            v2f fb = *(const v2f*)(wl + frag_off + (kc + 1) * 4);
            acc1 = __builtin_amdgcn_wmma_f32_16x16x4_f32(
                false, fb, false, fb, (short)0, acc1, false, false);
        }
        __syncthreads();
    }

    // ---- workgroup reduction of the 16x16 Gram fragments, then global atomics.
    // C layout: VGPR j, lane L -> G[m][n], m = j + 8*(L/16), n = L%16.
#pragma unroll
    for (int j = 0; j < 8; ++j) {
        const int m = j + 8 * (lane >> 4);
        const int n = lane & 15;
        lds[wave * 256 + m * 16 + n] = acc0[j] + acc1[j];
    }
    __syncthreads();

#pragma unroll
    for (int e = threadIdx.x; e < 256; e += BLOCK) {
        float sum = 0.f;
#pragma unroll
        for (int w = 0; w < WAVES; ++w) sum += lds[w * 256 + e];
        atomicAdd(&ws[b * 256 + e], sum);
    }
}

// ---------------------------------------------------------------- epilogue
__global__ void SNNROE_finalize(const float* __restrict__ ws, float* __restrict__ out) {
    const int b = threadIdx.x;
    if (b >= B_DIM) return;
    const float* G = ws + b * 256;
    const float n_last = sqrtf(G[15 * 16 + 15] + 1e-5f);
    float csum = 0.f;
#pragma unroll
    for (int t = 0; t < T_DIM; ++t) {
        const float nt = sqrtf(G[t * 16 + t] + 1e-5f);
        csum += G[t * 16 + 15] / (nt * n_last);
    }
    const float cs = csum * (1.0f / (float)T_DIM);
    out[b] = 1.0f / (cs + 1e-5f);
}

// ---------------------------------------------------------------- launcher
extern "C" void kernel_launch(void* const* d_in, const int* in_sizes, int n_in,
                              void* d_out, int out_size, void* d_ws, size_t ws_size,
                              hipStream_t stream) {
    const float* x  = (const float*)d_in[0];   // [16,32,64,32,32] f32
    // d_in[1] ("mem") is unused by the reference.
    float* out = (float*)d_out;                // [1,32] f32
    float* ws  = (float*)d_ws;                 // 32*256 f32 Gram accumulator

    SNNROE_zero<<<dim3((B_DIM * 256 + 255) / 256), 256, 0, stream>>>(ws);
    SNNROE_gram<<<dim3(B_DIM, WGS_PER_B), BLOCK, 0, stream>>>(x, ws);
    SNNROE_finalize<<<1, 32, 0, stream>>>(ws, out);
}